// ReLU_47940424958601
// MI455X (gfx1250) — compile-verified
//
#include <hip/hip_runtime.h>
#include <hip/hip_bf16.h>
#include <cstdint>

typedef __attribute__((ext_vector_type(4))) float        v4f;
typedef __attribute__((ext_vector_type(4))) unsigned int v4u;
typedef __attribute__((ext_vector_type(4))) int          v4i;
typedef __attribute__((ext_vector_type(8))) int          v8i;

// Elements (floats) moved per TENSOR_STORE_FROM_LDS op: 32 KB LDS tile.
#define TDM_OP_ELEMS 8192u

#if !__has_builtin(__builtin_amdgcn_tensor_store_from_lds)
#define TDM_MODE 0
#elif __has_include(<hip/amd_detail/amd_gfx1250_TDM.h>)
#define TDM_MODE 6   // amdgpu-toolchain / therock: 6-arg builtin
#else
#define TDM_MODE 5   // ROCm 7.2 clang-22: 5-arg builtin
#endif

// ---------------------------------------------------------------------------
// Kernel 1: bulk zero fill of d_out via the Tensor Data Mover.
// One wave per workgroup. Each op DMA-copies a 32 KB LDS tile of zeros to a
// contiguous 32 KB global slice. Descriptor layout per CDNA5 ISA ch.8.
// ---------------------------------------------------------------------------
__global__ void __launch_bounds__(32) tdm_zero_fill(float* __restrict__ out,
                                                    unsigned long long n_ops) {
#if TDM_MODE
  __shared__ __align__(128) float zbuf[TDM_OP_ELEMS];
  {
    v4f z = {0.0f, 0.0f, 0.0f, 0.0f};
    v4f* zb = (v4f*)zbuf;
    for (unsigned i = threadIdx.x; i < TDM_OP_ELEMS / 4u; i += blockDim.x)
      zb[i] = z;
  }
  __syncthreads();  // LDS zeros visible before TDM reads them

  const unsigned lds_off = (unsigned)(size_t)&zbuf[0];  // low 32b of flat ptr = LDS byte offset
  const unsigned TD0 = TDM_OP_ELEMS;                    // 1-D tile length in elements

  for (unsigned long long t = blockIdx.x; t < n_ops; t += gridDim.x) {
    unsigned long long ga = (unsigned long long)(size_t)out
                          + t * (unsigned long long)(TDM_OP_ELEMS * 4u);
    // D# group 0: count=1 (valid), lds_addr, global_addr[56:0], type=2
    v4u g0 = { 1u,
               lds_off,
               (unsigned)ga,
               (unsigned)((ga >> 32) & 0x1FFFFFFu) | (2u << 30) };
    // D# group 1: data_size=2 (4B), tensor_dim0=TD0, tensor_dim1=1,
    //             tile_dim0=TD0 (1-D tile), tensor_dim0_stride=TD0
    v8i g1 = { (int)(2u << 16),
               (int)((TD0 & 0xFFFFu) << 16),
               (int)(((TD0 >> 16) & 0xFFFFu) | (1u << 16)),
               (int)((TD0 & 0xFFFFu) << 16),
               0,
               (int)TD0,
               0, 0 };
    // D# group 2: tensor_dim2=1, tensor_dim3=1 (unused dims, benign)
    v4i g2 = { 1, 1, 0, 0 };
    // D# group 3: tensor_dim4=1 at bits[79:48]
    v4i g3 = { 0, (int)(1u << 16), 0, 0 };
#if TDM_MODE == 5
    __builtin_amdgcn_tensor_store_from_lds(g0, g1, g2, g3, 0);
#else
    v8i g4 = {0, 0, 0, 0, 0, 0, 0, 0};
    __builtin_amdgcn_tensor_store_from_lds(g0, g1, g2, g3, g4, 0);
#endif
  }
  __builtin_amdgcn_s_wait_tensorcnt(0);
#else
  // Fallback: plain wide non-temporal stores (global_store_b128 th:NT).
  v4f z = {0.0f, 0.0f, 0.0f, 0.0f};
  for (unsigned long long t = blockIdx.x; t < n_ops; t += gridDim.x) {
    v4f* dst = (v4f*)(out + t * (unsigned long long)TDM_OP_ELEMS);
    for (unsigned i = threadIdx.x; i < TDM_OP_ELEMS / 4u; i += blockDim.x)
      __builtin_nontemporal_store(z, dst + i);
  }
#endif
}

// ---------------------------------------------------------------------------
// Kernel 2: zero the scalar tail (out_size % TDM_OP_ELEMS elements).
// ---------------------------------------------------------------------------
__global__ void zero_range(float* __restrict__ out,
                           unsigned long long start, unsigned long long total) {
  unsigned long long i =
      start + (unsigned long long)blockIdx.x * blockDim.x + threadIdx.x;
  if (i < total) out[i] = 0.0f;
}

// ---------------------------------------------------------------------------
// Kernel 3: per-node relaxation values scattered into the pre-zeroed layout:
//   [conc_low (n) | conc_up (n) | A_low (n+1)^2 | A_up (n+1)^2]
// ---------------------------------------------------------------------------
__global__ void relu_relax_values(const float* __restrict__ lower,
                                  const float* __restrict__ upper,
                                  const float* __restrict__ alphas,
                                  float* __restrict__ out, int n) {
  int i = blockIdx.x * blockDim.x + threadIdx.x;
  long long np1 = (long long)n + 1;
  float* conc_low = out;
  float* conc_up  = out + n;
  float* A_low    = out + 2LL * n;
  float* A_up     = A_low + np1 * np1;

  if (i < n) {
    float l = lower[i];
    float u = upper[i];
    float a = fminf(fmaxf(alphas[i], 0.0f), 1.0f);
    bool active   = (u > 0.0f) && (l >= 0.0f);
    bool unstable = (u > 0.0f) && (l < 0.0f);
    float denom = u - l;
    float lam = u / ((denom == 0.0f) ? 1.0f : denom);

    float dl = active ? 1.0f : (unstable ? a : 0.0f);
    float du = active ? 1.0f : (unstable ? lam : 0.0f);
    float bu = unstable ? (-lam * l) : 0.0f;

    conc_low[i] = active ? l : (unstable ? a * l : 0.0f);
    conc_up[i]  = (u > 0.0f) ? u : 0.0f;

    long long d = (long long)i * np1 + i;
    A_low[d] = dl;
    A_up[d]  = du;
    A_up[(long long)n * np1 + i] = bu;   // last row of the upper relaxation
  } else if (i == n) {
    A_low[np1 * np1 - 1] = 1.0f;
    A_up[np1 * np1 - 1]  = 1.0f;
  }
}

// ---------------------------------------------------------------------------
extern "C" void kernel_launch(void* const* d_in, const int* in_sizes, int n_in,
                              void* d_out, int out_size, void* d_ws, size_t ws_size,
                              hipStream_t stream) {
  (void)n_in; (void)d_ws; (void)ws_size;
  const float* lower  = (const float*)d_in[0];
  const float* upper  = (const float*)d_in[1];
  const float* alphas = (const float*)d_in[2];
  float* out = (float*)d_out;
  int n = in_sizes[0];  // 8192

  unsigned long long total = (unsigned long long)(long long)out_size;  // floats
  unsigned long long n_ops = total / (unsigned long long)TDM_OP_ELEMS;
  unsigned long long tail_start = n_ops * (unsigned long long)TDM_OP_ELEMS;

  if (n_ops) {
    unsigned int grid =
        (unsigned int)((n_ops < 2048ull) ? n_ops : 2048ull);
    tdm_zero_fill<<<grid, 32, 0, stream>>>(out, n_ops);
  }
  if (tail_start < total) {
    unsigned long long tail = total - tail_start;
    zero_range<<<(unsigned int)((tail + 255ull) / 256ull), 256, 0, stream>>>(
        out, tail_start, total);
  }
  relu_relax_values<<<(unsigned int)((n + 1 + 255) / 256), 256, 0, stream>>>(
      lower, upper, alphas, out, n);
}